// SileroVADCoreML_20186346291661
// MI455X (gfx1250) — compile-verified
//
#include <hip/hip_runtime.h>
#include <hip/hip_bf16.h>
#include <math.h>

typedef __attribute__((ext_vector_type(2))) float v2f;
typedef __attribute__((ext_vector_type(8))) float v8f;

#define DEV static __device__ __forceinline__

DEV float sigm_(float x) { return 1.0f / (1.0f + __expf(-x)); }
DEV int   imin_(int a, int b) { return a < b ? a : b; }

// ---------------------------------------------------------------------------
// Generic zero-fill
// ---------------------------------------------------------------------------
__global__ void k_zero(float* __restrict__ p, int n)
{
    const int i = blockIdx.x * blockDim.x + threadIdx.x;
    if (i < n) p[i] = 0.0f;
}

// ---------------------------------------------------------------------------
// Pack kernels (all tiny, L2-resident; remove ALL branches from GEMM K-loops)
// ---------------------------------------------------------------------------
// Reflect-padded audio: padA[b*640 + j]
__global__ void k_pad(const float* __restrict__ audio, float* __restrict__ padA)
{
    const int i = blockIdx.x * blockDim.x + threadIdx.x;   // 8192*640
    const int b = i / 640, j = i - b * 640;
    padA[i] = (j < 576) ? audio[b * 576 + j] : audio[b * 576 + 1150 - j];
}

// enc_w1 padded to stride 388 (zero tail): w1p[n*388 + k]
__global__ void k_packw1(const float* __restrict__ w1, float* __restrict__ w1p)
{
    const int i = blockIdx.x * blockDim.x + threadIdx.x;   // 128*388 = 49664
    const int n = i / 388, k = i - n * 388;
    w1p[i] = (k < 387) ? w1[n * 387 + k] : 0.0f;
}

// LSTM weights packed: wcat[n*256 + k], n = gate*128 + j, k = x||h
__global__ void k_packw(const float* __restrict__ wi_, const float* __restrict__ wf_,
                        const float* __restrict__ wg_, const float* __restrict__ wo_,
                        const float* __restrict__ ui_, const float* __restrict__ uf_,
                        const float* __restrict__ ug_, const float* __restrict__ uo_,
                        float* __restrict__ wcat)
{
    const int i = blockIdx.x * blockDim.x + threadIdx.x;   // 512*256 = 131072
    const int n = i >> 8, k = i & 255;
    const int g = n >> 7, j = n & 127;
    const float* Wp = (g == 0) ? wi_ : (g == 1) ? wf_ : (g == 2) ? wg_ : wo_;
    const float* Up = (g == 0) ? ui_ : (g == 1) ? uf_ : (g == 2) ? ug_ : uo_;
    wcat[i] = (k < 128) ? Wp[j * 128 + k] : Up[j * 128 + k - 128];
}

// h into the x||h activation buffer: xh[b*256 + 128 + j]
__global__ void k_packxh(const float* __restrict__ hin, float* __restrict__ xh)
{
    const int i = blockIdx.x * blockDim.x + threadIdx.x;   // 8192*128
    const int b = i >> 7, j = i & 127;
    xh[b * 256 + 128 + j] = hin[i];
}

// ---------------------------------------------------------------------------
// Kernel 1: STFT GEMM  (M=32768 (b,t), N=258, K=256) -> stft[r*272 + ch]
// ---------------------------------------------------------------------------
__global__ void k_stft(const float* __restrict__ padA,
                       const float* __restrict__ fb,
                       float* __restrict__ stft)
{
    const int wave  = (blockIdx.x * blockDim.x + threadIdx.x) >> 5;
    const int mtile = wave / 17, ntile = wave % 17;        // NT = 17
    const int mbase = mtile * 16, nbase = ntile * 16;
    const int lane  = threadIdx.x & 31, half = lane >> 4, lm = lane & 15;

    const int r = mbase + lm;
    const float* arow = padA + (r >> 2) * 640 + (r & 3) * 128;
    // Columns >= 258 are never stored; clamp pointer so garbage is harmless.
    const float* brow = fb + imin_(nbase + lm, 257) * 256;

    v8f acc = {};
    for (int s = 0; s < 64; ++s) {
        const int k0 = s * 4 + 2 * half;
        v2f a, bb;
        a.x  = arow[k0];  a.y  = arow[k0 + 1];
        bb.x = brow[k0];  bb.y = brow[k0 + 1];
        acc = __builtin_amdgcn_wmma_f32_16x16x4_f32(false, a, false, bb,
                                                    (short)0, acc, false, false);
    }
#pragma unroll
    for (int v = 0; v < 8; ++v) {
        const int row = mbase + v + 8 * half;
        const int col = nbase + lm;
        if (col < 258) stft[row * 272 + col] = acc[v];
    }
}

// ---------------------------------------------------------------------------
// Kernel 2: magnitude with zero halo: magp[(b*129+c)*6 + slot], slot=t+1
// ---------------------------------------------------------------------------
__global__ void k_mag(const float* __restrict__ stft, float* __restrict__ magp)
{
    const int i = blockIdx.x * blockDim.x + threadIdx.x;   // 8192*129*6
    if (i >= 8192 * 129 * 6) return;
    const int bc = i / 6, slot = i - bc * 6;
    if (slot == 0 || slot == 5) { magp[i] = 0.0f; return; }
    const int b = bc / 129, c = bc - b * 129;
    const int t = slot - 1;
    const int row = (b * 4 + t) * 272;
    const float re = stft[row + c];
    const float im = stft[row + c + 129];
    magp[i] = sqrtf(re * re + im * im);
}

// ---------------------------------------------------------------------------
// Kernel 3: enc1  (M=32768, N=128, K=388 padded)  -> e1p[(b*6+t+1)*128+col]
// ---------------------------------------------------------------------------
__global__ void k_enc1(const float* __restrict__ magp,
                       const float* __restrict__ w1p, const float* __restrict__ b1,
                       float* __restrict__ e1p)
{
    const int wave  = (blockIdx.x * blockDim.x + threadIdx.x) >> 5;
    const int mtile = wave >> 3, ntile = wave & 7;         // NT = 8
    const int mbase = mtile * 16, nbase = ntile * 16;
    const int lane  = threadIdx.x & 31, half = lane >> 4, lm = lane & 15;
    const int r = mbase + lm, b = r >> 2, t = r & 3;
    const float* mrow = magp + b * 129 * 6 + t;            // + c*6 + dt
    const float* brow = w1p + (nbase + lm) * 388;

    v8f acc = {};
    for (int s = 0; s < 97; ++s) {
        const int k0 = s * 4 + 2 * half;
        const int k1 = k0 + 1;
        // k=387 hits the zero tail of w1p; clamp c so A stays in bounds.
        const int c0 = imin_(k0 / 3, 128), d0 = k0 - 3 * (k0 / 3);
        const int c1 = imin_(k1 / 3, 128), d1 = k1 - 3 * (k1 / 3);
        v2f a, bb;
        a.x  = mrow[c0 * 6 + d0];
        a.y  = mrow[c1 * 6 + d1];
        bb.x = brow[k0];
        bb.y = brow[k1];
        acc = __builtin_amdgcn_wmma_f32_16x16x4_f32(false, a, false, bb,
                                                    (short)0, acc, false, false);
    }
#pragma unroll
    for (int v = 0; v < 8; ++v) {
        const int rr = mbase + v + 8 * half;
        const int bb2 = rr >> 2, tt = rr & 3;
        const int col = nbase + lm;
        e1p[(bb2 * 6 + tt + 1) * 128 + col] = fmaxf(acc[v] + b1[col], 0.0f);
    }
}

// ---------------------------------------------------------------------------
// Kernel 4: enc2  (M=16384 (b,t2), N=64, K=384) -> e2p[(b*4+t2+1)*64+col]
// ---------------------------------------------------------------------------
__global__ void k_enc2(const float* __restrict__ e1p,
                       const float* __restrict__ w2, const float* __restrict__ b2,
                       float* __restrict__ e2p)
{
    const int wave  = (blockIdx.x * blockDim.x + threadIdx.x) >> 5;
    const int mtile = wave >> 2, ntile = wave & 3;         // NT = 4
    const int mbase = mtile * 16, nbase = ntile * 16;
    const int lane  = threadIdx.x & 31, half = lane >> 4, lm = lane & 15;
    const int r = mbase + lm, b = r >> 1, t2 = r & 1;
    const float* erow = e1p + (b * 6 + 2 * t2) * 128;      // slot = 2*t2+dt
    const float* brow = w2 + (nbase + lm) * 384;

    v8f acc = {};
    for (int s = 0; s < 96; ++s) {
        const int k0 = s * 4 + 2 * half;
        const int k1 = k0 + 1;
        const int c0 = k0 / 3, d0 = k0 - 3 * c0;
        const int c1 = k1 / 3, d1 = k1 - 3 * c1;
        v2f a, bb;
        a.x  = erow[d0 * 128 + c0];
        a.y  = erow[d1 * 128 + c1];
        bb.x = brow[k0];
        bb.y = brow[k1];
        acc = __builtin_amdgcn_wmma_f32_16x16x4_f32(false, a, false, bb,
                                                    (short)0, acc, false, false);
    }
#pragma unroll
    for (int v = 0; v < 8; ++v) {
        const int rr = mbase + v + 8 * half;
        const int bb2 = rr >> 1, tt = rr & 1;
        const int col = nbase + lm;
        e2p[(bb2 * 4 + tt + 1) * 64 + col] = fmaxf(acc[v] + b2[col], 0.0f);
    }
}

// ---------------------------------------------------------------------------
// Kernel 5: enc3  (M=8192, N=64, K=192) -> e3[b*64+col]
// ---------------------------------------------------------------------------
__global__ void k_enc3(const float* __restrict__ e2p,
                       const float* __restrict__ w3, const float* __restrict__ b3,
                       float* __restrict__ e3)
{
    const int wave  = (blockIdx.x * blockDim.x + threadIdx.x) >> 5;
    const int mtile = wave >> 2, ntile = wave & 3;         // NT = 4
    const int mbase = mtile * 16, nbase = ntile * 16;
    const int lane  = threadIdx.x & 31, half = lane >> 4, lm = lane & 15;
    const float* erow = e2p + (mbase + lm) * 4 * 64;       // slot = dt
    const float* brow = w3 + (nbase + lm) * 192;

    v8f acc = {};
    for (int s = 0; s < 48; ++s) {
        const int k0 = s * 4 + 2 * half;
        const int k1 = k0 + 1;
        const int c0 = k0 / 3, d0 = k0 - 3 * c0;
        const int c1 = k1 / 3, d1 = k1 - 3 * c1;
        v2f a, bb;
        a.x  = erow[d0 * 64 + c0];
        a.y  = erow[d1 * 64 + c1];
        bb.x = brow[k0];
        bb.y = brow[k1];
        acc = __builtin_amdgcn_wmma_f32_16x16x4_f32(false, a, false, bb,
                                                    (short)0, acc, false, false);
    }
#pragma unroll
    for (int v = 0; v < 8; ++v) {
        const int row = mbase + v + 8 * half;
        const int col = nbase + lm;
        e3[row * 64 + col] = fmaxf(acc[v] + b3[col], 0.0f);
    }
}

// ---------------------------------------------------------------------------
// Kernel 6: enc4  (M=8192, N=128, K=64; center tap) -> xh[b*256+col]
// ---------------------------------------------------------------------------
__global__ void k_enc4(const float* __restrict__ e3,
                       const float* __restrict__ w4, const float* __restrict__ b4,
                       float* __restrict__ xh)
{
    const int wave  = (blockIdx.x * blockDim.x + threadIdx.x) >> 5;
    const int mtile = wave >> 3, ntile = wave & 7;         // NT = 8
    const int mbase = mtile * 16, nbase = ntile * 16;
    const int lane  = threadIdx.x & 31, half = lane >> 4, lm = lane & 15;
    const float* erow = e3 + (mbase + lm) * 64;
    const float* brow = w4 + (nbase + lm) * 192 + 1;       // dt=1 taps

    v8f acc = {};
    for (int s = 0; s < 16; ++s) {
        const int k0 = s * 4 + 2 * half;
        v2f a, bb;
        a.x  = erow[k0];
        a.y  = erow[k0 + 1];
        bb.x = brow[k0 * 3];
        bb.y = brow[k0 * 3 + 3];
        acc = __builtin_amdgcn_wmma_f32_16x16x4_f32(false, a, false, bb,
                                                    (short)0, acc, false, false);
    }
#pragma unroll
    for (int v = 0; v < 8; ++v) {
        const int row = mbase + v + 8 * half;
        const int col = nbase + lm;
        xh[row * 256 + col] = fmaxf(acc[v] + b4[col], 0.0f);
    }
}

// ---------------------------------------------------------------------------
// Kernel 7: LSTM gate GEMM  (M=8192, N=512, K=256, fully packed operands)
// ---------------------------------------------------------------------------
__global__ void k_gates(const float* __restrict__ xh, const float* __restrict__ wcat,
                        float* __restrict__ gates)
{
    const int wave  = (blockIdx.x * blockDim.x + threadIdx.x) >> 5;
    const int mtile = wave >> 5, ntile = wave & 31;        // NT = 32
    const int mbase = mtile * 16, nbase = ntile * 16;
    const int lane  = threadIdx.x & 31, half = lane >> 4, lm = lane & 15;
    const float* arow = xh + (mbase + lm) * 256;
    const float* brow = wcat + (nbase + lm) * 256;

    v8f acc = {};
    for (int s = 0; s < 64; ++s) {
        const int k0 = s * 4 + 2 * half;
        v2f a, bb;
        a.x  = arow[k0];  a.y  = arow[k0 + 1];
        bb.x = brow[k0];  bb.y = brow[k0 + 1];
        acc = __builtin_amdgcn_wmma_f32_16x16x4_f32(false, a, false, bb,
                                                    (short)0, acc, false, false);
    }
#pragma unroll
    for (int v = 0; v < 8; ++v) {
        const int row = mbase + v + 8 * half;
        const int col = nbase + lm;
        gates[row * 512 + col] = acc[v];
    }
}

// ---------------------------------------------------------------------------
// Kernel 8: LSTM cell elementwise + final dot/sigmoid (one block per row)
// ---------------------------------------------------------------------------
__global__ void k_lstm(const float* __restrict__ gates, const float* __restrict__ cin,
                       const float* __restrict__ bxi, const float* __restrict__ bxf,
                       const float* __restrict__ bxg, const float* __restrict__ bxo,
                       const float* __restrict__ bhi, const float* __restrict__ bhf,
                       const float* __restrict__ bhg, const float* __restrict__ bho,
                       const float* __restrict__ convw, const float* __restrict__ convb,
                       float* __restrict__ out_prob, float* __restrict__ out_h,
                       float* __restrict__ out_c)
{
    const int b = blockIdx.x;
    const int j = threadIdx.x;                             // 0..127
    const float* G = gates + b * 512;

    const float iv = sigm_(G[j]       + bxi[j] + bhi[j]);
    const float fv = sigm_(G[128 + j] + bxf[j] + bhf[j]);
    const float gv = tanhf(G[256 + j] + bxg[j] + bhg[j]);
    const float ov = sigm_(G[384 + j] + bxo[j] + bho[j]);

    const float co = fv * cin[b * 128 + j] + iv * gv;
    const float ho = ov * tanhf(co);
    out_h[b * 128 + j] = ho;
    out_c[b * 128 + j] = co;

    __shared__ float red[128];
    red[j] = fmaxf(ho, 0.0f) * convw[j];
    __syncthreads();
#pragma unroll
    for (int s2 = 64; s2 > 0; s2 >>= 1) {
        if (j < s2) red[j] += red[j + s2];
        __syncthreads();
    }
    if (j == 0) out_prob[b] = sigm_(red[0] + convb[0]);
}

// ---------------------------------------------------------------------------
extern "C" void kernel_launch(void* const* d_in, const int* in_sizes, int n_in,
                              void* d_out, int out_size, void* d_ws, size_t ws_size,
                              hipStream_t stream)
{
    const float* audio = (const float*)d_in[0];
    const float* h_in  = (const float*)d_in[1];
    const float* c_in  = (const float*)d_in[2];
    const float* fb    = (const float*)d_in[3];
    const float* w1 = (const float*)d_in[4];  const float* b1 = (const float*)d_in[5];
    const float* w2 = (const float*)d_in[6];  const float* b2 = (const float*)d_in[7];
    const float* w3 = (const float*)d_in[8];  const float* b3 = (const float*)d_in[9];
    const float* w4 = (const float*)d_in[10]; const float* b4 = (const float*)d_in[11];
    const float* wi = (const float*)d_in[12]; const float* wf = (const float*)d_in[13];
    const float* wg = (const float*)d_in[14]; const float* wo = (const float*)d_in[15];
    const float* ui = (const float*)d_in[16]; const float* uf = (const float*)d_in[17];
    const float* ug = (const float*)d_in[18]; const float* uo = (const float*)d_in[19];
    const float* bxi = (const float*)d_in[20]; const float* bxf = (const float*)d_in[21];
    const float* bxg = (const float*)d_in[22]; const float* bxo = (const float*)d_in[23];
    const float* bhi = (const float*)d_in[24]; const float* bhf = (const float*)d_in[25];
    const float* bhg = (const float*)d_in[26]; const float* bho = (const float*)d_in[27];
    const float* convw = (const float*)d_in[28];
    const float* convb = (const float*)d_in[29];

    // ---- workspace layout (floats), region reuse; peak ~62 MB ----
    float* ws    = (float*)d_ws;
    float* stft  = ws;                 //  8,912,896  (dead after k_mag)
    float* padA  = ws + 8950000;       //  5,242,880  (dead after k_stft)
    float* magp  = ws + 8950000;       //  6,340,608  (over dead padA; dead after enc1)
    float* e1p   = ws;                 //  6,291,456  (over dead stft; dead after enc2)
    float* e2p   = ws + 6291456;       //  2,097,152  (dead after enc3)
    float* e3    = ws;                 //    524,288  (over dead e1p)
    float* xh    = ws + 600000;        //  2,097,152
    float* gates = ws + 8950000;       //  4,194,304  (over dead magp)
    float* wcat  = ws + 15300000;      //    131,072
    float* w1p   = ws + 15500000;      //     49,664  (total 15,549,664 fl = 62.2 MB)

    float* out      = (float*)d_out;
    float* out_prob = out;                      // 8192
    float* out_h    = out + 8192;               // 8192*128
    float* out_c    = out + 8192 + 8192 * 128;  // 8192*128

    // weight packing (independent of activations)
    k_packw1<<<194,  256, 0, stream>>>(w1, w1p);
    k_packw <<<512,  256, 0, stream>>>(wi, wf, wg, wo, ui, uf, ug, uo, wcat);

    // pipeline (GEMMs: 256 thr = 8 wave32, one 16x16 f32-WMMA tile per wave)
    k_pad   <<<20480, 256, 0, stream>>>(audio, padA);
    k_stft  <<<4352,  256, 0, stream>>>(padA, fb, stft);        // 2048*17 tiles
    k_mag   <<<24768, 256, 0, stream>>>(stft, magp);
    k_zero  <<<24576, 256, 0, stream>>>(e1p, 6291456);
    k_enc1  <<<2048,  256, 0, stream>>>(magp, w1p, b1, e1p);    // 2048*8 tiles
    k_zero  <<<8192,  256, 0, stream>>>(e2p, 2097152);
    k_enc2  <<<512,   256, 0, stream>>>(e1p, w2, b2, e2p);      // 1024*4 tiles
    k_enc3  <<<256,   256, 0, stream>>>(e2p, w3, b3, e3);       // 512*4 tiles
    k_packxh<<<4096,  256, 0, stream>>>(h_in, xh);
    k_enc4  <<<512,   256, 0, stream>>>(e3, w4, b4, xh);        // 512*8 tiles
    k_gates <<<2048,  256, 0, stream>>>(xh, wcat, gates);       // 512*32 tiles
    k_lstm  <<<8192,  128, 0, stream>>>(gates, c_in, bxi, bxf, bxg, bxo,
                                        bhi, bhf, bhg, bho, convw, convb,
                                        out_prob, out_h, out_c);
}